// FinalADRModel_65317862637944
// MI455X (gfx1250) — compile-verified
//
#include <hip/hip_runtime.h>
#include <math.h>

#define NUM_NODES  100000
#define NUM_EDGES  3200000
#define EMBED_DIM  32
#define HIDDEN_DIM 64
#define LAB_DIM    9
#define BATCH      16384

typedef __attribute__((ext_vector_type(2))) float v2f;
typedef __attribute__((ext_vector_type(8))) float v8f;

// ---------------------------------------------------------------------------
// Degree / norm kernels
// ---------------------------------------------------------------------------
__global__ void init_deg_kernel(float* __restrict__ deg) {
    int i = blockIdx.x * 256 + threadIdx.x;
    if (i < NUM_NODES) deg[i] = 1.0f;  // self-loop
}

__global__ void deg_scatter_kernel(const int* __restrict__ dst, float* __restrict__ deg) {
    int e = blockIdx.x * 256 + threadIdx.x;
    if (e < NUM_EDGES) atomicAdd(&deg[dst[e]], 1.0f);
}

__global__ void dinv_kernel(float* __restrict__ deg) {
    int i = blockIdx.x * 256 + threadIdx.x;
    if (i < NUM_NODES) deg[i] = rsqrtf(deg[i]);  // deg >= 1 always
}

// ---------------------------------------------------------------------------
// Generic fp32 WMMA GEMM: out[nrows x 64] = X[nrows x K] @ W[K x 64]
//   optional per-row scale (dinv), optional bias, optional relu,
//   optional duplicate store (agg init = hs for GCN self-loop term).
// Block = 256 threads = 8 waves; each wave computes a 16x64 tile via
// 4x V_WMMA_F32_16X16X4_F32 per K-step of 4.
//
// W is staged in LDS as interleaved K-pairs: sW[p*64 + col] = {W[2p][col],
// W[2p+1][col]} so each B operand is one contiguous ds_load_b64 straight
// into an even-aligned VGPR pair (no v_mov reassembly).
// ---------------------------------------------------------------------------
__global__ __launch_bounds__(256)
void gemm64_wmma_kernel(const float* __restrict__ X, const float* __restrict__ W,
                        const float* __restrict__ bias, const float* __restrict__ rowscale,
                        float* __restrict__ out, float* __restrict__ out2,
                        int nrows, int K, int relu) {
    __shared__ float2 sW[48 * 64];  // max K = 96 -> 48 K-pairs x 64 cols = 24KB
    const int tid = threadIdx.x;
    const int npairs = (K >> 1) * 64;
    for (int i = tid; i < npairs; i += 256) {
        int p = i >> 6, col = i & 63;
        sW[i] = make_float2(W[(size_t)(2 * p) * 64 + col],
                            W[(size_t)(2 * p + 1) * 64 + col]);
    }
    __syncthreads();

    const int wave = tid >> 5;
    const int lane = tid & 31;
    const int half = lane >> 4;   // 0: lanes 0-15, 1: lanes 16-31
    const int ml   = lane & 15;

    const int row_base = blockIdx.x * 128 + wave * 16;
    int r = row_base + ml;
    int rl = r < nrows ? r : nrows - 1;          // clamp loads, guard stores
    const float* xr = X + (size_t)rl * (size_t)K;

    v8f acc0 = {}, acc1 = {}, acc2 = {}, acc3 = {};

    for (int k0 = 0; k0 < K; k0 += 4) {
        // A 16x4 f32 layout: lanes 0-15 hold K=k0,k0+1; lanes 16-31 hold K=k0+2,k0+3
        const int ka = k0 + half * 2;
        v2f a;
        a.x = xr[ka];
        a.y = xr[ka + 1];

        // B 4x16 tiles: pair index = ka/2; one b64 LDS load per operand
        const float2* wp = &sW[(size_t)((k0 >> 1) + half) * 64];
        v2f b0 = *(const v2f*)&wp[ 0 + ml];
        v2f b1 = *(const v2f*)&wp[16 + ml];
        v2f b2 = *(const v2f*)&wp[32 + ml];
        v2f b3 = *(const v2f*)&wp[48 + ml];

        acc0 = __builtin_amdgcn_wmma_f32_16x16x4_f32(false, a, false, b0, (short)0, acc0, false, false);
        acc1 = __builtin_amdgcn_wmma_f32_16x16x4_f32(false, a, false, b1, (short)0, acc1, false, false);
        acc2 = __builtin_amdgcn_wmma_f32_16x16x4_f32(false, a, false, b2, (short)0, acc2, false, false);
        acc3 = __builtin_amdgcn_wmma_f32_16x16x4_f32(false, a, false, b3, (short)0, acc3, false, false);
    }

    // C layout: VGPR v -> M = v + 8*half, N = ml (+16*n tile)
    #pragma unroll
    for (int v = 0; v < 8; ++v) {
        const int rr = row_base + v + half * 8;
        if (rr < nrows) {
            const float s = rowscale ? rowscale[rr] : 1.0f;
            float vals[4] = { acc0[v], acc1[v], acc2[v], acc3[v] };
            #pragma unroll
            for (int n = 0; n < 4; ++n) {
                const int col = n * 16 + ml;
                float val = vals[n] * s;
                if (bias) val += bias[col];
                if (relu) val = fmaxf(val, 0.0f);
                const size_t idx = (size_t)rr * 64 + col;
                out[idx] = val;
                if (out2) out2[idx] = val;
            }
        }
    }
}

// ---------------------------------------------------------------------------
// Edge scatter: agg[dst] += hs[src]; 16 threads per edge, 4 floats each.
// ---------------------------------------------------------------------------
__global__ void edge_scatter_kernel(const int* __restrict__ src, const int* __restrict__ dst,
                                    const float* __restrict__ hs, float* __restrict__ agg) {
    long long t = (long long)blockIdx.x * 256 + threadIdx.x;
    int e = (int)(t >> 4);
    if (e < NUM_EDGES) {
        int f = (int)(t & 15) * 4;
        int s = src[e], d = dst[e];
        const float4 v = *(const float4*)(hs + (size_t)s * 64 + f);
        float* p = agg + (size_t)d * 64 + f;
        atomicAdd(p + 0, v.x);
        atomicAdd(p + 1, v.y);
        atomicAdd(p + 2, v.z);
        atomicAdd(p + 3, v.w);
    }
}

// out[i][f] = relu?( dinv[i]*agg[i][f] + bias[f] )
__global__ void gcn_finalize_kernel(const float* __restrict__ agg, const float* __restrict__ dinv,
                                    const float* __restrict__ bias, float* __restrict__ out, int relu) {
    int t = blockIdx.x * 256 + threadIdx.x;
    if (t < NUM_NODES * 64) {
        int i = t >> 6, f = t & 63;
        float v = dinv[i] * agg[t] + bias[f];
        if (relu) v = fmaxf(v, 0.0f);
        out[t] = v;
    }
}

// ---------------------------------------------------------------------------
// Lab MLP: relu(lab @ W1 + b1) @ W2 + b2, per-thread row, weights in LDS
// ---------------------------------------------------------------------------
__global__ __launch_bounds__(256)
void lab_mlp_kernel(const float* __restrict__ lab, const float* __restrict__ W1,
                    const float* __restrict__ b1, const float* __restrict__ W2,
                    const float* __restrict__ b2, float* __restrict__ lab_embed) {
    __shared__ float sW1[LAB_DIM * 32], sb1[32], sW2[32 * 32], sb2[32];
    const int tid = threadIdx.x;
    for (int i = tid; i < LAB_DIM * 32; i += 256) sW1[i] = W1[i];
    for (int i = tid; i < 32 * 32;      i += 256) sW2[i] = W2[i];
    if (tid < 32) { sb1[tid] = b1[tid]; sb2[tid] = b2[tid]; }
    __syncthreads();

    int r = blockIdx.x * 256 + tid;
    if (r < BATCH) {
        float in[LAB_DIM];
        #pragma unroll
        for (int i = 0; i < LAB_DIM; ++i) in[i] = lab[(size_t)r * LAB_DIM + i];
        float h[32];
        #pragma unroll
        for (int j = 0; j < 32; ++j) {
            float a = sb1[j];
            #pragma unroll
            for (int i = 0; i < LAB_DIM; ++i) a += in[i] * sW1[i * 32 + j];
            h[j] = fmaxf(a, 0.0f);
        }
        for (int j = 0; j < 32; ++j) {
            float a = sb2[j];
            #pragma unroll
            for (int k = 0; k < 32; ++k) a += h[k] * sW2[k * 32 + j];
            lab_embed[(size_t)r * 32 + j] = a;
        }
    }
}

// combined[r][0:64] = nodeOut[drug_ids[r]], combined[r][64:96] = lab_embed[r]
__global__ void build_combined_kernel(const float* __restrict__ nodeOut,
                                      const float* __restrict__ lab_embed,
                                      const int* __restrict__ drug_ids,
                                      float* __restrict__ combined) {
    int t = blockIdx.x * 256 + threadIdx.x;
    if (t < BATCH * 96) {
        int r = t / 96, c = t - r * 96;
        float v = (c < 64) ? nodeOut[(size_t)drug_ids[r] * 64 + c]
                           : lab_embed[(size_t)r * 32 + (c - 64)];
        combined[t] = v;
    }
}

// logits = h_fus @ fus_W2 + fus_b2 ; sigmoid
__global__ void head_kernel(const float* __restrict__ h, const float* __restrict__ W2,
                            const float* __restrict__ b2, float* __restrict__ out) {
    int r = blockIdx.x * 256 + threadIdx.x;
    if (r < BATCH) {
        float s = b2[0];
        const float* hr = h + (size_t)r * 64;
        #pragma unroll
        for (int k = 0; k < 64; ++k) s += hr[k] * W2[k];
        out[r] = 1.0f / (1.0f + expf(-s));
    }
}

// ---------------------------------------------------------------------------
// Launch
// ---------------------------------------------------------------------------
extern "C" void kernel_launch(void* const* d_in, const int* in_sizes, int n_in,
                              void* d_out, int out_size, void* d_ws, size_t ws_size,
                              hipStream_t stream) {
    const int*   edge_index = (const int*)d_in[0];
    const int*   drug_ids   = (const int*)d_in[1];
    const float* lab_feat   = (const float*)d_in[2];
    const float* emb        = (const float*)d_in[3];
    const float* W1         = (const float*)d_in[4];
    const float* b1         = (const float*)d_in[5];
    const float* W2         = (const float*)d_in[6];
    const float* b2         = (const float*)d_in[7];
    const float* lab_W1     = (const float*)d_in[8];
    const float* lab_b1     = (const float*)d_in[9];
    const float* lab_W2     = (const float*)d_in[10];
    const float* lab_b2     = (const float*)d_in[11];
    const float* fus_W1     = (const float*)d_in[12];
    const float* fus_b1     = (const float*)d_in[13];
    const float* fus_W2     = (const float*)d_in[14];
    const float* fus_b2     = (const float*)d_in[15];
    float* out = (float*)d_out;

    const int* e_src = edge_index;
    const int* e_dst = edge_index + NUM_EDGES;

    // Workspace layout (floats), with overlaid reuse of dead regions:
    float* ws = (float*)d_ws;
    const size_t NFEAT = (size_t)NUM_NODES * 64;            // 6,400,000
    float* dinv   = ws;                                     // 100000
    float* hs     = ws + 102400;                            // NFEAT   (later: nodeOut)
    float* agg    = hs + NFEAT;                             // NFEAT   (later: combined)
    float* xbuf   = agg + NFEAT;                            // NFEAT   (later: h_fus @0, lab_embed @+2M)
    float* nodeOut   = hs;
    float* combined  = agg;
    float* h_fus     = xbuf;
    float* lab_embed = xbuf + 2000000;

    const int nodeBlocks  = (NUM_NODES + 255) / 256;
    const int edgeBlocks  = (NUM_EDGES + 255) / 256;
    const int scatBlocks  = (int)(((long long)NUM_EDGES * 16 + 255) / 256);
    const int featBlocks  = (int)((NFEAT + 255) / 256);
    const int gemmBlocksN = (NUM_NODES + 127) / 128;
    const int gemmBlocksB = (BATCH + 127) / 128;
    const int batchBlocks = (BATCH + 255) / 256;

    // ---- degree / norm (shared by both convs) ----
    init_deg_kernel<<<nodeBlocks, 256, 0, stream>>>(dinv);
    deg_scatter_kernel<<<edgeBlocks, 256, 0, stream>>>(e_dst, dinv);
    dinv_kernel<<<nodeBlocks, 256, 0, stream>>>(dinv);

    // ---- conv1: hs = (emb @ W1) * dinv[row]; agg = hs (self-loop) ----
    gemm64_wmma_kernel<<<gemmBlocksN, 256, 0, stream>>>(
        emb, W1, nullptr, dinv, hs, agg, NUM_NODES, EMBED_DIM, 0);
    edge_scatter_kernel<<<scatBlocks, 256, 0, stream>>>(e_src, e_dst, hs, agg);
    gcn_finalize_kernel<<<featBlocks, 256, 0, stream>>>(agg, dinv, b1, xbuf, 1);

    // ---- conv2 ----
    gemm64_wmma_kernel<<<gemmBlocksN, 256, 0, stream>>>(
        xbuf, W2, nullptr, dinv, hs, agg, NUM_NODES, HIDDEN_DIM, 0);
    edge_scatter_kernel<<<scatBlocks, 256, 0, stream>>>(e_src, e_dst, hs, agg);
    gcn_finalize_kernel<<<featBlocks, 256, 0, stream>>>(agg, dinv, b2, nodeOut, 0);

    // ---- batch path ----
    lab_mlp_kernel<<<batchBlocks, 256, 0, stream>>>(lab_feat, lab_W1, lab_b1, lab_W2, lab_b2, lab_embed);
    build_combined_kernel<<<(BATCH * 96 + 255) / 256, 256, 0, stream>>>(nodeOut, lab_embed, drug_ids, combined);
    gemm64_wmma_kernel<<<gemmBlocksB, 256, 0, stream>>>(
        combined, fus_W1, fus_b1, nullptr, h_fus, nullptr, BATCH, 96, 1);
    head_kernel<<<batchBlocks, 256, 0, stream>>>(h_fus, fus_W2, fus_b2, out);
}